// TorchROIPoolingLayer_89335319757589
// MI455X (gfx1250) — compile-verified
//
#include <hip/hip_runtime.h>
#include <cstdint>
#include <cstddef>

#ifndef __has_builtin
#define __has_builtin(x) 0
#endif

// Problem constants (match the reference)
#define N_C   512
#define N_H   50
#define N_W   50
#define P_H   7
#define P_W   7
#define NPOS  49            // 7*7 pooled positions
#define SSCALE 0.0625f
#define PLANE (N_H * N_W)   // 2500 floats per (b,c) plane

typedef float vfloat4 __attribute__((ext_vector_type(4)));

// One workgroup (256 threads, 8 wave32) per ROI.
__global__ __launch_bounds__(256) void roi_align_gfx1250(
    const float* __restrict__ feat,   // (4, 512, 50, 50)
    const float* __restrict__ rois,   // (N, 5) = {batch, x1, y1, x2, y2}
    float* __restrict__ out,          // (N, 512, 7, 7)
    int n_rois)
{
    __shared__ int4   soff[NPOS];     // 4 gather offsets per pooled position
    __shared__ float4 swt[NPOS];      // 4 bilinear weights per pooled position
    __shared__ float  sroi[8];        // staged ROI record (async global->LDS)

    const int roi = blockIdx.x;
    if (roi >= n_rois) return;

    // ---- Stage the 5 ROI floats into LDS on the CDNA5 async global->LDS
    // path (tracked by ASYNCcnt, completed with s_wait_asynccnt). Five lanes
    // of wave 0 each move one dword.
    if (threadIdx.x < 5) {
        const float* gp = rois + (size_t)roi * 5 + threadIdx.x;
        unsigned lds_off = (unsigned)(uintptr_t)(&sroi[threadIdx.x]);
        asm volatile("global_load_async_to_lds_b32 %0, %1, off"
                     :: "v"(lds_off), "v"(gp)
                     : "memory");
    }
#if __has_builtin(__builtin_amdgcn_s_wait_asynccnt)
    __builtin_amdgcn_s_wait_asynccnt(0);
#else
    asm volatile("s_wait_asynccnt 0x0" ::: "memory");
#endif
    __syncthreads();

    // ---- Per-ROI geometry (uniform across the block)
    const int   b  = (int)sroi[0];
    const float x1 = floorf(sroi[1] * SSCALE);
    const float y1 = floorf(sroi[2] * SSCALE);
    const float x2 = floorf(sroi[3] * SSCALE);
    const float y2 = floorf(sroi[4] * SSCALE);

    // Threads 0..48 precompute the 49 positions' gather offsets + weights.
    if (threadIdx.x < NPOS) {
        const int p = threadIdx.x;
        const int i = p / P_W;            // pooled row
        const int j = p - i * P_W;        // pooled col

        const float xs  = x1 + (x2 - x1) * (float)j / (float)(P_W - 1);
        const float ys  = y1 + (y2 - y1) * (float)i / (float)(P_H - 1);
        const float x0f = floorf(xs);
        const float y0f = floorf(ys);
        const float fx  = xs - x0f;
        const float fy  = ys - y0f;

        int x0  = min(max((int)x0f, 0), N_W - 1);
        int x1i = min(x0 + 1, N_W - 1);
        int y0  = min(max((int)y0f, 0), N_H - 1);
        int y1i = min(y0 + 1, N_H - 1);

        soff[p] = make_int4(y0  * N_W + x0, y0  * N_W + x1i,
                            y1i * N_W + x0, y1i * N_W + x1i);
        swt[p]  = make_float4((1.0f - fy) * (1.0f - fx),
                              (1.0f - fy) * fx,
                              fy * (1.0f - fx),
                              fy * fx);
    }
    __syncthreads();

    // ---- Main loop: each thread produces 4 consecutive outputs and stores
    // them as one non-temporal b128. Per-ROI output block is 512*49 = 25088
    // floats (divisible by 4; base 16B-aligned): aligned, fully coalesced,
    // and rinsed through L2 (output is write-once, feat stays resident).
    const float* __restrict__ fb = feat + (size_t)b * (N_C * PLANE);
    float* __restrict__ ob = out + (size_t)roi * (N_C * NPOS);

    const unsigned total = N_C * NPOS;           // 25088
    for (unsigned q = threadIdx.x * 4u; q < total; q += 256u * 4u) {
        // One division per round; quad elements wrap incrementally (4 < 49).
        const unsigned c0 = q / NPOS;
        const unsigned p0 = q - c0 * NPOS;

        // Pull the channel plane ~one round ahead toward the WGP cache.
        const unsigned cpre = c0 + 21u;
        if (cpre < N_C)
            __builtin_prefetch(fb + (size_t)cpre * PLANE, 0, 0);

        float r[4];
#pragma unroll
        for (int e = 0; e < 4; ++e) {
            unsigned p = p0 + (unsigned)e;
            unsigned c = c0;
            if (p >= NPOS) { p -= NPOS; c += 1u; }   // at most one wrap
            const float* __restrict__ pc = fb + (size_t)c * PLANE;
            const int4   o = soff[p];   // ds_load_b128
            const float4 w = swt[p];    // ds_load_b128
            r[e] = w.x * pc[o.x] + w.y * pc[o.y]
                 + w.z * pc[o.z] + w.w * pc[o.w];
        }
        vfloat4 v = { r[0], r[1], r[2], r[3] };
        __builtin_nontemporal_store(v, reinterpret_cast<vfloat4*>(ob + q));
    }
}

extern "C" void kernel_launch(void* const* d_in, const int* in_sizes, int n_in,
                              void* d_out, int out_size, void* d_ws, size_t ws_size,
                              hipStream_t stream)
{
    const float* feat = (const float*)d_in[0];   // (4,512,50,50) float32
    const float* rois = (const float*)d_in[1];   // (N,5) float32
    float* out = (float*)d_out;                  // (N,512,7,7) float32

    const int n_rois = in_sizes[1] / 5;          // 4000
    dim3 grid((unsigned)n_rois), block(256);
    hipLaunchKernelGGL(roi_align_gfx1250, grid, block, 0, stream,
                       feat, rois, out, n_rois);
}